// FullyVectorizedSimilaritySmoothing_62766652064369
// MI455X (gfx1250) — compile-verified
//
#include <hip/hip_runtime.h>
#include <math.h>
#include <stdint.h>

typedef __attribute__((ext_vector_type(2))) float v2f;
typedef __attribute__((ext_vector_type(8))) float v8f;

#define B_ 8
#define S_ 2048
#define H_ 256
#define P_ 64
#define NQ_ 20
#define N_ (B_ * S_)          // 16384 rows
#define HC_ 256               // concatenated hidden (128 sim | 128 conf)

// CDNA5 async global->LDS copy (ASYNCcnt-tracked). The LDS destination is a
// per-lane VGPR address, so this performs the packed-layout permutation while
// copying, with no VGPR round trip.
__device__ __forceinline__ void async_f32_to_lds(uint32_t lds_byte_off,
                                                 const float* gptr) {
    asm volatile("global_load_async_to_lds_b32 %0, %1, off"
                 :: "v"(lds_byte_off), "v"(gptr)
                 : "memory");
}
__device__ __forceinline__ void wait_async_zero() {
    asm volatile("s_wait_asynccnt 0x0" ::: "memory");
}

// ---------------------------------------------------------------------------
// Kernel 1: fused dual-MLP layer.  X[N,256] x Wcat[256,256] -> relu -> dot w2
// Each wave computes one 16-row M-tile across all 16 N-tiles using
// V_WMMA_F32_16X16X4_F32 (full fp32 precision).  Weights staged in LDS in
// K-slabs of 64 (64KB) via async-to-LDS copies, packed in (k/2, j, k&1) order
// so a ds_load_2addr_b64 yields the exact B-fragment layout.
// ---------------------------------------------------------------------------
__global__ __launch_bounds__(256)
void mlp_wmma_kernel(const float* __restrict__ X,
                     const float* __restrict__ sim_w1, const float* __restrict__ sim_b1,
                     const float* __restrict__ sim_w2, const float* __restrict__ sim_b2,
                     const float* __restrict__ conf_w1, const float* __restrict__ conf_b1,
                     const float* __restrict__ conf_w2, const float* __restrict__ conf_b2,
                     float* __restrict__ simA, float* __restrict__ confA)
{
    __shared__ float wlds[64 * 256];   // one K-slab of packed Wcat: 64KB

    const int tid  = threadIdx.x;
    const int lane = tid & 31;
    const int wave = tid >> 5;
    const int jc   = lane & 15;        // N / M index within tile
    const int half = lane >> 4;        // K-group (0: K=0,1  1: K=2,3)

    const int mtile = blockIdx.x * 8 + wave;   // 128 blocks * 8 waves = 1024 tiles
    const int row0  = mtile * 16;
    const int arow  = row0 + jc;

    const uint32_t lds_base = (uint32_t)(uintptr_t)&wlds[0];

    v8f acc[16] = {};

    for (int slab = 0; slab < 4; ++slab) {
        __syncthreads();               // previous slab fully consumed
        // async copy of 64x256 weight slab, packed for b64 B-frag loads
        for (int idx = tid; idx < 64 * 256; idx += 256) {
            const int kk = idx >> 8;           // 0..63 (slab-local k)
            const int j  = idx & 255;
            const int k  = slab * 64 + kk;
            const float* src = (j < 128) ? (sim_w1  + k * 128 + j)
                                         : (conf_w1 + k * 128 + (j - 128));
            const uint32_t dst = lds_base +
                4u * (uint32_t)((kk >> 1) * 512 + j * 2 + (kk & 1));
            async_f32_to_lds(dst, src);
        }
        wait_async_zero();             // my async writes landed in LDS
        __syncthreads();               // everyone's writes visible

        for (int k0 = 0; k0 < 64; k0 += 4) {
            const int kB = k0 + 2 * half;      // slab-local, even
            // preload all 16 B fragments into distinct registers so the LDS
            // loads overlap the WMMA stream instead of serializing on waits
            v2f bfr[16];
#pragma unroll
            for (int t = 0; t < 16; ++t) {
                const int j = t * 16 + jc;
                bfr[t] = *(const v2f*)(&wlds[(kB >> 1) * 512 + j * 2]);
            }
            // A fragment: lane holds X[arow, kA], X[arow, kA+1]
            const int kA = slab * 64 + k0 + 2 * half;
            const v2f a = *(const v2f*)(X + (size_t)arow * H_ + kA);
#pragma unroll
            for (int t = 0; t < 16; ++t) {
                acc[t] = __builtin_amdgcn_wmma_f32_16x16x4_f32(
                    false, a, false, bfr[t], (short)0, acc[t], false, false);
            }
        }
    }

    // Epilogue: bias + relu + scale by w2, reduce over 256 hidden cols.
    // Lane holds rows (v + 8*half) for column subset { t*16 + jc }.
    float ssim[8] = {0.f,0.f,0.f,0.f,0.f,0.f,0.f,0.f};
    float sconf[8] = {0.f,0.f,0.f,0.f,0.f,0.f,0.f,0.f};
#pragma unroll
    for (int t = 0; t < 16; ++t) {
        const int col = t * 16 + jc;
        const float b1 = (col < 128) ? sim_b1[col] : conf_b1[col - 128];
        const float w2 = (col < 128) ? sim_w2[col] : conf_w2[col - 128];
#pragma unroll
        for (int v = 0; v < 8; ++v) {
            float h = acc[t][v] + b1;
            h = h > 0.f ? h : 0.f;
            const float contrib = h * w2;
            if (t < 8) ssim[v] += contrib; else sconf[v] += contrib;
        }
    }
    // reduce across the 16 lanes of each half (columns mod 16)
#pragma unroll
    for (int v = 0; v < 8; ++v) {
#pragma unroll
        for (int off = 8; off >= 1; off >>= 1) {
            ssim[v]  += __shfl_xor(ssim[v],  off, 16);
            sconf[v] += __shfl_xor(sconf[v], off, 16);
        }
    }
    if (jc == 0) {
        const float b2s = sim_b2[0];
        const float b2c = conf_b2[0];
#pragma unroll
        for (int v = 0; v < 8; ++v) {
            const int row = row0 + half * 8 + v;
            simA[row] = ssim[v] + b2s;
            const float cpre = sconf[v] + b2c;
            confA[row] = 1.f / (1.f + __expf(-cpre));
        }
    }
}

// ---------------------------------------------------------------------------
// Kernel 2: per (batch, class) softmax aggregation.
// grid = B*NQ blocks, 256 threads.
// ---------------------------------------------------------------------------
__global__ __launch_bounds__(256)
void agg_kernel(const float* __restrict__ param_x,
                const float* __restrict__ simA,
                const int* __restrict__ questions,
                float* __restrict__ smoothed)
{
    const int b = blockIdx.x / NQ_;
    const int q = blockIdx.x % NQ_;
    const int base = b * S_;
    const int tid = threadIdx.x;

    __shared__ float red[256];
    __shared__ int redi[256];
    __shared__ float s_m, s_denom;
    __shared__ int s_cnt;

    // pass 1: masked max + count
    float lmax = -3.4e38f;
    int lcnt = 0;
    for (int i = tid; i < S_; i += 256) {
        if (questions[base + i] == q) {
            lmax = fmaxf(lmax, simA[base + i]);
            lcnt++;
        }
    }
    red[tid] = lmax; redi[tid] = lcnt;
    __syncthreads();
    for (int off = 128; off >= 1; off >>= 1) {
        if (tid < off) {
            red[tid] = fmaxf(red[tid], red[tid + off]);
            redi[tid] += redi[tid + off];
        }
        __syncthreads();
    }
    if (tid == 0) {
        const int cnt = redi[0];
        float m = red[0];
        if (cnt < S_) m = fmaxf(m, 0.f);   // zeros from masked-out entries
        if (cnt == 0) m = 0.f;
        s_m = m; s_cnt = cnt;
    }
    __syncthreads();
    const float m = s_m;
    const int cnt = s_cnt;

    // pass 2: weighted sums.  p = tid&63, 4 row-lanes stride over i.
    const int p = tid & 63;
    const int rl = tid >> 6;
    float accW = 0.f, accP = 0.f, accT = 0.f, accD = 0.f;
    for (int i = rl; i < S_; i += 4) {
        const float px = param_x[(size_t)(base + i) * P_ + p];
        accT += px;
        if (questions[base + i] == q) {
            const float e = __expf(simA[base + i] - m);
            accW += e * px;
            accP += px;
            if (p == 0) accD += e;
        }
    }
    __syncthreads();
    float W = 0.f, Pm = 0.f, T = 0.f;
    red[tid] = accW; __syncthreads();
    if (tid < 64) W = red[tid] + red[tid + 64] + red[tid + 128] + red[tid + 192];
    __syncthreads();
    red[tid] = accP; __syncthreads();
    if (tid < 64) Pm = red[tid] + red[tid + 64] + red[tid + 128] + red[tid + 192];
    __syncthreads();
    red[tid] = accT; __syncthreads();
    if (tid < 64) T = red[tid] + red[tid + 64] + red[tid + 128] + red[tid + 192];
    __syncthreads();
    red[tid] = accD; __syncthreads();
    if (tid == 0)
        s_denom = red[0] + red[64] + red[128] + red[192]
                + (float)(S_ - cnt) * __expf(-m);
    __syncthreads();

    if (tid < 64) {
        const float sm = (W + __expf(-m) * (T - Pm)) / s_denom;
        smoothed[((size_t)b * NQ_ + q) * P_ + tid] = sm;
    }
}

// ---------------------------------------------------------------------------
// Kernel 3: out = c*param + (1-c)*smoothed[b, q[n], p]
// ---------------------------------------------------------------------------
__global__ __launch_bounds__(256)
void blend_kernel(const float* __restrict__ param_x,
                  const float* __restrict__ confA,
                  const int* __restrict__ questions,
                  const float* __restrict__ smoothed,
                  float* __restrict__ out)
{
    const int idx = blockIdx.x * 256 + threadIdx.x;
    if (idx >= N_ * P_) return;
    const int n = idx >> 6;
    const int p = idx & 63;
    const int b = n >> 11;            // S_ = 2048
    const float c = confA[n];
    const int q = questions[n];
    out[idx] = c * param_x[idx]
             + (1.f - c) * smoothed[((size_t)b * NQ_ + q) * P_ + p];
}

extern "C" void kernel_launch(void* const* d_in, const int* in_sizes, int n_in,
                              void* d_out, int out_size, void* d_ws, size_t ws_size,
                              hipStream_t stream)
{
    const float* feature_x = (const float*)d_in[0];
    const float* param_x   = (const float*)d_in[1];
    const float* sim_w1    = (const float*)d_in[2];
    const float* sim_b1    = (const float*)d_in[3];
    const float* sim_w2    = (const float*)d_in[4];
    const float* sim_b2    = (const float*)d_in[5];
    const float* conf_w1   = (const float*)d_in[6];
    const float* conf_b1   = (const float*)d_in[7];
    const float* conf_w2   = (const float*)d_in[8];
    const float* conf_b2   = (const float*)d_in[9];
    const int*   questions = (const int*)d_in[10];
    float* out = (float*)d_out;

    float* ws       = (float*)d_ws;
    float* simA     = ws;                 // [N_]
    float* confA    = ws + N_;            // [N_]
    float* smoothed = ws + 2 * N_;        // [B_*NQ_*P_]

    mlp_wmma_kernel<<<N_ / (16 * 8), 256, 0, stream>>>(
        feature_x, sim_w1, sim_b1, sim_w2, sim_b2,
        conf_w1, conf_b1, conf_w2, conf_b2, simA, confA);

    agg_kernel<<<B_ * NQ_, 256, 0, stream>>>(param_x, simA, questions, smoothed);

    blend_kernel<<<(N_ * P_ + 255) / 256, 256, 0, stream>>>(
        param_x, confA, questions, smoothed, out);
}